// Microscope_8083128451457
// MI455X (gfx1250) — compile-verified
//
#include <hip/hip_runtime.h>

// ---- Problem constants (match reference) ----
#define NPZ 11
#define NPY 21
#define NPX 21
#define PSF_VOL (NPZ * NPY * NPX)   // 4851
#define DIM_B 2
#define DIM_C 4
#define DIM_D 48
#define DIM_H 256
#define DIM_W 256
#define SCALE_F 10000.0f

typedef unsigned int u32x4 __attribute__((ext_vector_type(4)));
typedef int          i32x8 __attribute__((ext_vector_type(8)));
typedef int          i32x4 __attribute__((ext_vector_type(4)));

// ---------------------------------------------------------------------------
// Zero-init d_out (harness poisons it with 0xAA before timing).
// ---------------------------------------------------------------------------
__global__ void zero_out_kernel(float* __restrict__ out, int n4, int n) {
    int i = blockIdx.x * blockDim.x + threadIdx.x;
    int stride = gridDim.x * blockDim.x;
    float4 z = make_float4(0.f, 0.f, 0.f, 0.f);
    for (int k = i; k < n4; k += stride)
        reinterpret_cast<float4*>(out)[k] = z;
    // scalar tail (out_size is divisible by 4 here, but stay safe)
    for (int k = n4 * 4 + i; k < n; k += stride)
        out[k] = 0.f;
}

// Bounds-checked LDS template read: OOB -> 0 (zero-pad semantics of reference)
__device__ __forceinline__ float ldT(const float* s, int z, int y, int x) {
    bool ok = ((unsigned)z < (unsigned)NPZ) &
              ((unsigned)y < (unsigned)NPY) &
              ((unsigned)x < (unsigned)NPX);
    int idx = ok ? ((z * NPY + y) * NPX + x) : 0;
    float v = s[idx];
    return ok ? v : 0.0f;
}

// ---------------------------------------------------------------------------
// One emitter per wave32. PSF template staged in LDS via TDM async load.
// Per voxel: 8 LDS taps, trilinear combine, clamp, scale, one fp32 L2 atomic.
// ---------------------------------------------------------------------------
__global__ __launch_bounds__(256) void scatter_psf_kernel(
    const int*   __restrict__ eb,  const int*   __restrict__ ec,
    const int*   __restrict__ ez,  const int*   __restrict__ ey,
    const int*   __restrict__ ex,
    const float* __restrict__ xos, const float* __restrict__ yos,
    const float* __restrict__ zos, const float* __restrict__ ival,
    const float* __restrict__ psf, const float* __restrict__ sc_fac,
    float* __restrict__ out, int n_emitters)
{
    __shared__ float sPsf[PSF_VOL];

    const int tid  = threadIdx.x;
    const int lane = tid & 31;
    const int wave = tid >> 5;
    const int wavesPerBlock = blockDim.x >> 5;

    // ---- Stage the shared PSF template into LDS ----
#if defined(__has_builtin) && __has_builtin(__builtin_amdgcn_tensor_load_to_lds)
    if (wave == 0) {
        // Tensor DMA descriptor: 1-D copy, 4851 x 4B, global -> LDS offset 0.
        unsigned long long ga = (unsigned long long)psf;
        u32x4 g0;
        g0[0] = 1u;                                   // count=1 (valid user D#)
        g0[1] = 0u;                                   // lds_addr = 0 (sPsf is sole LDS alloc)
        g0[2] = (unsigned)(ga & 0xFFFFFFFFu);         // global_addr[31:0]
        g0[3] = (unsigned)((ga >> 32) & 0x1FFFFFFu)   // global_addr[56:32]
              | (2u << 30);                           // type = 2 ("image")
        i32x8 g1;
        g1[0] = (int)(2u << 16);                      // workgroup_mask=0, data_size=2 (4B)
        g1[1] = (int)((unsigned)(PSF_VOL & 0xFFFF) << 16);   // tensor_dim0[15:0]
        g1[2] = (int)((PSF_VOL >> 16) | (1u << 16)); // tensor_dim0[31:16], tensor_dim1=1
        g1[3] = (int)((unsigned)PSF_VOL << 16);      // tensor_dim1[31:16]=0, tile_dim0=4851
        g1[4] = 1;                                    // tile_dim1=1, tile_dim2=0
        g1[5] = PSF_VOL;                              // tensor_dim0_stride[31:0]
        g1[6] = 0;                                    // stride0[47:32], stride1[15:0]
        g1[7] = 0;                                    // stride1[47:16]
        i32x4 gz4 = {0, 0, 0, 0};                     // groups 2/3 unused (<=2D tensor)
        i32x8 gz8 = {0, 0, 0, 0, 0, 0, 0, 0};
        __builtin_amdgcn_tensor_load_to_lds(g0, g1, gz4, gz4, gz8, 0);
#if __has_builtin(__builtin_amdgcn_s_wait_tensorcnt)
        __builtin_amdgcn_s_wait_tensorcnt(0);
#else
        asm volatile("s_wait_tensorcnt 0" ::: "memory");
#endif
    }
#else
    for (int i = tid; i < PSF_VOL; i += blockDim.x)
        sPsf[i] = psf[i];
#endif

    // The TDM engine wrote sPsf, but no IR-visible store exists; escape the
    // pointer through an asm memory clobber so the optimizer cannot fold the
    // LDS loads below to undef (observed: ds=0 without this).
    {
        void* p = (void*)sPsf;
        asm volatile("" : "+v"(p) : : "memory");
    }
    __syncthreads();

    const float sc = sc_fac[0] * SCALE_F;

    for (int e0 = blockIdx.x * wavesPerBlock + wave; e0 < n_emitters;
         e0 += gridDim.x * wavesPerBlock) {
        // e is wave-uniform: force SGPR so emitter params become scalar loads
        const int e = __builtin_amdgcn_readfirstlane(e0);

        const int   b  = eb[e], c = ec[e];
        const int   z0 = ez[e], y0 = ey[e], x0 = ex[e];
        const float dz = zos[e], dy = yos[e], dx = xos[e];
        const float amp = fmaxf(ival[e], 0.0f) * sc;

        // sampled at q = i - d ; i0 = i + floor(-d) ; w1 = (-d) - floor(-d)
        const float fz = floorf(-dz); const int kz = (int)fz;
        const float wz1 = -dz - fz,   wz0 = 1.0f - wz1;
        const float fy = floorf(-dy); const int ky = (int)fy;
        const float wy1 = -dy - fy,   wy0 = 1.0f - wy1;
        const float fx = floorf(-dx); const int kx = (int)fx;
        const float wx1 = -dx - fx,   wx0 = 1.0f - wx1;

        const int bcBase = (b * DIM_C + c) * DIM_D;

        // x fastest-varying across lanes -> consecutive atomic addresses
        for (int v = lane; v < PSF_VOL; v += 32) {
            const int iz  = v / (NPY * NPX);
            const int rem = v - iz * (NPY * NPX);
            const int iy  = rem / NPX;
            const int ix  = rem - iy * NPX;

            const int oz = z0 - (NPZ / 2) + iz;
            const int oy = y0 - (NPY / 2) + iy;
            const int ox = x0 - (NPX / 2) + ix;
            const bool inb = ((unsigned)oz < (unsigned)DIM_D) &
                             ((unsigned)oy < (unsigned)DIM_H) &
                             ((unsigned)ox < (unsigned)DIM_W);
            if (!inb) continue;   // reference: mode='drop'

            const int tz = iz + kz, ty = iy + ky, tx = ix + kx;
            const float t000 = ldT(sPsf, tz,     ty,     tx);
            const float t001 = ldT(sPsf, tz,     ty,     tx + 1);
            const float t010 = ldT(sPsf, tz,     ty + 1, tx);
            const float t011 = ldT(sPsf, tz,     ty + 1, tx + 1);
            const float t100 = ldT(sPsf, tz + 1, ty,     tx);
            const float t101 = ldT(sPsf, tz + 1, ty,     tx + 1);
            const float t110 = ldT(sPsf, tz + 1, ty + 1, tx);
            const float t111 = ldT(sPsf, tz + 1, ty + 1, tx + 1);

            const float s0 = wy0 * (wx0 * t000 + wx1 * t001) +
                             wy1 * (wx0 * t010 + wx1 * t011);
            const float s1 = wy0 * (wx0 * t100 + wx1 * t101) +
                             wy1 * (wx0 * t110 + wx1 * t111);
            const float val = fmaxf(wz0 * s0 + wz1 * s1, 0.0f) * amp;

            const size_t oidx =
                ((size_t)(bcBase + oz) * DIM_H + (size_t)oy) * DIM_W + (size_t)ox;
            unsafeAtomicAdd(out + oidx, val);   // global_atomic_add_f32, no return
        }
    }
}

// ---------------------------------------------------------------------------
extern "C" void kernel_launch(void* const* d_in, const int* in_sizes, int n_in,
                              void* d_out, int out_size, void* d_ws, size_t ws_size,
                              hipStream_t stream) {
    (void)n_in; (void)d_ws; (void)ws_size;

    const int*   eb   = (const int*)  d_in[0];
    const int*   ec   = (const int*)  d_in[1];
    const int*   ez   = (const int*)  d_in[2];
    const int*   ey   = (const int*)  d_in[3];
    const int*   ex   = (const int*)  d_in[4];
    const float* xos  = (const float*)d_in[5];
    const float* yos  = (const float*)d_in[6];
    const float* zos  = (const float*)d_in[7];
    const float* ival = (const float*)d_in[8];
    const float* psf  = (const float*)d_in[9];
    const float* scf  = (const float*)d_in[10];
    float*       out  = (float*)d_out;
    const int n = in_sizes[0];

    // 1) zero the 100.7 MB output volume (L2-resident afterwards)
    const int n4 = out_size >> 2;
    zero_out_kernel<<<2048, 256, 0, stream>>>(out, n4, out_size);

    // 2) scatter: one emitter per wave32, 8 waves (256 threads) per block
    const int wavesPerBlock = 8;
    const int blocks = (n + wavesPerBlock - 1) / wavesPerBlock;
    scatter_psf_kernel<<<blocks, 256, 0, stream>>>(
        eb, ec, ez, ey, ex, xos, yos, zos, ival, psf, scf, out, n);
}